// SelfAttentionLayer_40046275067911
// MI455X (gfx1250) — compile-verified
//
#include <hip/hip_runtime.h>

typedef __attribute__((ext_vector_type(2))) float v2f;
typedef __attribute__((ext_vector_type(8))) float v8f;
typedef __attribute__((ext_vector_type(4))) unsigned int u32x4;
typedef __attribute__((ext_vector_type(8))) int i32x8;
typedef __attribute__((ext_vector_type(4))) int i32x4;

#define E_DIM 1536
#define NTOK  32768   // 8 * 4096 tokens
#define TM 128
#define TN 128
#define TK 32
#define LSTR 36       // LDS row stride in floats (32 data + 4 pad)

// Use the Tensor Data Mover when the toolchain exposes it (6-arg clang-23 form).
#if defined(__gfx1250__) && __has_builtin(__builtin_amdgcn_tensor_load_to_lds) && \
    __has_builtin(__builtin_amdgcn_s_wait_tensorcnt)
#define USE_TDM 1
#else
#define USE_TDM 0
#endif

#if USE_TDM
// Return the 32-bit LDS byte offset of a generic pointer into __shared__ space.
__device__ __forceinline__ unsigned lds_byte_offset(void* p) {
    return (unsigned)(unsigned long long)(__attribute__((address_space(3))) void*)p;
}

// Issue one TDM descriptor moving a [tile_rows x 32] fp32 tile (row stride 1536
// elements in global) into LDS at lds_off, with HW padding of 4 dwords after
// every 32 dwords -> LDS row stride 36 floats (matches the fragment readers).
// Wave-level op: call from one wave only; tracked by TENSORcnt.
__device__ __forceinline__ void tdm_load_tile(const float* gtile, unsigned lds_off,
                                              unsigned tensor_rows, unsigned tile_rows) {
    unsigned long long ga = (unsigned long long)gtile;
    u32x4 g0 = {
        1u,                                              // count=1, user descriptor
        lds_off,                                         // lds_addr (bytes)
        (unsigned)(ga & 0xffffffffu),                    // global_addr[31:0]
        (unsigned)((ga >> 32) & 0x01ffffffu) | (2u << 30) // global_addr[56:32] | type=2
    };
    i32x8 g1 = {
        (int)((2u << 16) | (1u << 20) | (4u << 22) | (3u << 25)),
        // ^ data_size=4B, pad_enable, pad_interval=32 dwords, pad_amount=4 dwords
        (int)((1536u & 0xffffu) << 16),                  // tensor_dim0 = 1536 (lo16)
        (int)(((1536u >> 16) & 0xffffu) | ((tensor_rows & 0xffffu) << 16)),
        (int)(((tensor_rows >> 16) & 0xffffu) | (32u << 16)),   // | tile_dim0 = 32
        (int)(tile_rows & 0xffffu),                      // tile_dim1, tile_dim2=0
        (int)1536,                                       // tensor_dim0_stride lo32
        0, 0                                             // stride hi, dim1_stride (2D)
    };
    i32x4 gz4 = { 0, 0, 0, 0 };                          // groups 2/3 disabled (2D)
    i32x8 gz8 = { 0, 0, 0, 0, 0, 0, 0, 0 };              // extension group unused
    __builtin_amdgcn_tensor_load_to_lds(g0, g1, gz4, gz4, gz8, 0);
}
#endif

// Shared inner compute: one K-chunk of 32 on a 128x128 tile.
__device__ __forceinline__ void mma_chunk(const float* __restrict__ Ab,
                                          const float* __restrict__ Bb,
                                          int m_wave, int n_wave, int row16, int khalf,
                                          v8f acc[2][4]) {
    #pragma unroll
    for (int ks = 0; ks < TK; ks += 4) {
        const int col = ks + 2 * khalf;          // even -> 8B-aligned ds_load_b64
        v2f a[2], b[4];
        #pragma unroll
        for (int mi = 0; mi < 2; ++mi)
            a[mi] = *(const v2f*)&Ab[(m_wave + mi * 16 + row16) * LSTR + col];
        #pragma unroll
        for (int ni = 0; ni < 4; ++ni)
            b[ni] = *(const v2f*)&Bb[(n_wave + ni * 16 + row16) * LSTR + col];
        #pragma unroll
        for (int mi = 0; mi < 2; ++mi)
            #pragma unroll
            for (int ni = 0; ni < 4; ++ni)
                acc[mi][ni] = __builtin_amdgcn_wmma_f32_16x16x4_f32(
                    false, a[mi], false, b[ni], (short)0, acc[mi][ni], false, false);
    }
}

// ---------------------------------------------------------------------------
// C[M,1536] = A[M,1536] . W[1536,1536]^T  (+ optional bias[n] + resid[m,n])
// fp32 end-to-end with V_WMMA_F32_16X16X4_F32. 256 threads (8 waves),
// tile 128x128, TK=32 staged in LDS. TDM path: double-buffered DMA tiles.
// ---------------------------------------------------------------------------
__global__ __launch_bounds__(256) void gemm_abt_kernel(
    const float* __restrict__ A,
    const float* __restrict__ W,
    float* __restrict__ C,
    const float* __restrict__ bias,    // may be null
    const float* __restrict__ resid,   // may be null
    int a_rows)                        // rows of A tensor (OOB clamp bound)
{
    __shared__ __align__(16) float As[2][TM * LSTR];
    __shared__ __align__(16) float Bs[2][TN * LSTR];

    const int tid   = threadIdx.x;
    const int lane  = tid & 31;
    const int wave  = tid >> 5;
    const int m0    = blockIdx.y * TM;
    const int n0    = blockIdx.x * TN;

    const int m_wave = (wave >> 1) * 32;
    const int n_wave = (wave & 1) * 64;
    const int row16  = lane & 15;
    const int khalf  = lane >> 4;

    v8f acc[2][4];
    v8f zero = {};
    #pragma unroll
    for (int i = 0; i < 2; ++i)
        #pragma unroll
        for (int j = 0; j < 4; ++j) acc[i][j] = zero;

#if USE_TDM
    // Prologue: DMA first K-chunk into buffer 0.
    if (wave == 0) {
        tdm_load_tile(A + (size_t)m0 * E_DIM, lds_byte_offset(&As[0][0]),
                      (unsigned)a_rows, TM);
        tdm_load_tile(W + (size_t)n0 * E_DIM, lds_byte_offset(&Bs[0][0]),
                      (unsigned)E_DIM, TN);
        __builtin_amdgcn_s_wait_tensorcnt(0);
    }
    __syncthreads();

    int buf = 0;
    for (int k0 = 0; k0 < E_DIM; k0 += TK) {
        // Kick DMA for next chunk into the other buffer while we compute.
        if (wave == 0 && (k0 + TK) < E_DIM) {
            tdm_load_tile(A + (size_t)m0 * E_DIM + (k0 + TK),
                          lds_byte_offset(&As[buf ^ 1][0]), (unsigned)a_rows, TM);
            tdm_load_tile(W + (size_t)n0 * E_DIM + (k0 + TK),
                          lds_byte_offset(&Bs[buf ^ 1][0]), (unsigned)E_DIM, TN);
        }
        mma_chunk(&As[buf][0], &Bs[buf][0], m_wave, n_wave, row16, khalf, acc);
        if (wave == 0) __builtin_amdgcn_s_wait_tensorcnt(0);
        __syncthreads();
        buf ^= 1;
    }
#else
    for (int k0 = 0; k0 < E_DIM; k0 += TK) {
        #pragma unroll
        for (int i = 0; i < 4; ++i) {
            const int idx = tid + i * 256;
            const int row = idx >> 3;
            const int c4  = idx & 7;
            const float4 av = *(const float4*)(A + (size_t)(m0 + row) * E_DIM + k0 + c4 * 4);
            const float4 bv = *(const float4*)(W + (size_t)(n0 + row) * E_DIM + k0 + c4 * 4);
            *(float4*)&As[0][row * LSTR + c4 * 4] = av;
            *(float4*)&Bs[0][row * LSTR + c4 * 4] = bv;
        }
        __syncthreads();
        if (k0 + TK < E_DIM) {
            const int row = tid >> 1;
            __builtin_prefetch(A + (size_t)(m0 + row) * E_DIM + k0 + TK, 0, 0);
            __builtin_prefetch(W + (size_t)(n0 + row) * E_DIM + k0 + TK, 0, 0);
        }
        mma_chunk(&As[0][0], &Bs[0][0], m_wave, n_wave, row16, khalf, acc);
        __syncthreads();
    }
#endif

    // C/D layout: VGPR r, lanes 0-15 -> M=r, lanes 16-31 -> M=r+8; N = lane%16.
    #pragma unroll
    for (int mi = 0; mi < 2; ++mi) {
        #pragma unroll
        for (int ni = 0; ni < 4; ++ni) {
            #pragma unroll
            for (int r = 0; r < 8; ++r) {
                const int m = m0 + m_wave + mi * 16 + r + 8 * khalf;
                const int n = n0 + n_wave + ni * 16 + row16;
                float v = acc[mi][ni][r];
                if (bias)  v += bias[n];
                if (resid) v += resid[(size_t)m * E_DIM + n];
                C[(size_t)m * E_DIM + n] = v;
            }
        }
    }
}

// ---------------------------------------------------------------------------
// Per-token head-mixing attention: Q,K,V rows of 1536 viewed as [12][128].
// E = Q.K^T / sqrt(1536); softmax rows; O = A.V. O overwrites Xq in place.
// ---------------------------------------------------------------------------
__global__ __launch_bounds__(256) void attn_kernel(
    float* __restrict__ Xq,
    const float* __restrict__ Xk,
    const float* __restrict__ Xv)
{
    __shared__ float qs[E_DIM];
    __shared__ float kk[E_DIM];
    __shared__ float vv[E_DIM];
    __shared__ float att[12 * 12];

    const int tid = threadIdx.x;
    const size_t base = (size_t)blockIdx.x * E_DIM;

    for (int i = tid; i < E_DIM; i += 256) {
        qs[i] = Xq[base + i];
        kk[i] = Xk[base + i];
        vv[i] = Xv[base + i];
    }
    __syncthreads();

    if (tid < 144) {
        const int i = tid / 12, j = tid % 12;
        const float* qp = &qs[i * 128];
        const float* kp = &kk[j * 128];
        float s = 0.f;
        #pragma unroll 8
        for (int d = 0; d < 128; ++d) s += qp[d] * kp[d];
        att[tid] = s * 0.02551551815f;   // 1/sqrt(1536)
    }
    __syncthreads();

    if (tid < 12) {
        float mx = -1e30f;
        #pragma unroll
        for (int j = 0; j < 12; ++j) mx = fmaxf(mx, att[tid * 12 + j]);
        float ev[12], sum = 0.f;
        #pragma unroll
        for (int j = 0; j < 12; ++j) { ev[j] = __expf(att[tid * 12 + j] - mx); sum += ev[j]; }
        const float inv = 1.f / sum;
        #pragma unroll
        for (int j = 0; j < 12; ++j) att[tid * 12 + j] = ev[j] * inv;
    }
    __syncthreads();

    for (int idx = tid; idx < E_DIM; idx += 256) {
        const int q = idx >> 7, d = idx & 127;
        float o = 0.f;
        #pragma unroll
        for (int l = 0; l < 12; ++l) o += att[q * 12 + l] * vv[l * 128 + d];
        Xq[base + idx] = o;
    }
}

// ---------------------------------------------------------------------------
// In-place LayerNorm over the last dim (1536) with gamma/beta.
// ---------------------------------------------------------------------------
__global__ __launch_bounds__(256) void layernorm_kernel(
    float* __restrict__ out,
    const float* __restrict__ gamma,
    const float* __restrict__ beta)
{
    __shared__ float rs[256];
    __shared__ float rs2[256];
    const int tid = threadIdx.x;
    const size_t base = (size_t)blockIdx.x * E_DIM;

    float x[6], s = 0.f, s2 = 0.f;
    #pragma unroll
    for (int i = 0; i < 6; ++i) {
        x[i] = out[base + tid + i * 256];
        s += x[i];
        s2 += x[i] * x[i];
    }
    rs[tid] = s; rs2[tid] = s2;
    __syncthreads();
    for (int off = 128; off > 0; off >>= 1) {
        if (tid < off) { rs[tid] += rs[tid + off]; rs2[tid] += rs2[tid + off]; }
        __syncthreads();
    }
    const float mean = rs[0] * (1.f / 1536.f);
    const float var  = rs2[0] * (1.f / 1536.f) - mean * mean;
    const float inv  = rsqrtf(var + 1e-5f);
    #pragma unroll
    for (int i = 0; i < 6; ++i) {
        const int n = tid + i * 256;
        out[base + n] = (x[i] - mean) * inv * gamma[n] + beta[n];
    }
}

// ---------------------------------------------------------------------------
extern "C" void kernel_launch(void* const* d_in, const int* in_sizes, int n_in,
                              void* d_out, int out_size, void* d_ws, size_t ws_size,
                              hipStream_t stream) {
    const float* values = (const float*)d_in[0];
    const float* keys_  = (const float*)d_in[1];
    const float* query  = (const float*)d_in[2];
    const float* Wv     = (const float*)d_in[3];
    const float* Wk     = (const float*)d_in[4];
    const float* Wq     = (const float*)d_in[5];
    const float* Wo     = (const float*)d_in[6];
    const float* bo     = (const float*)d_in[7];
    const float* gamma  = (const float*)d_in[8];
    const float* beta   = (const float*)d_in[9];
    float* out = (float*)d_out;

    const size_t mat = (size_t)NTOK * E_DIM;
    float* Xq = (float*)d_ws;
    float* Xk = Xq + mat;
    float* Xv = Xk + mat;

    dim3 grid(E_DIM / TN, NTOK / TM);            // (12, 256)
    dim3 blk(256);

    gemm_abt_kernel<<<grid, blk, 0, stream>>>(query,  Wq, Xq, nullptr, nullptr, NTOK);
    gemm_abt_kernel<<<grid, blk, 0, stream>>>(keys_,  Wk, Xk, nullptr, nullptr, NTOK);
    gemm_abt_kernel<<<grid, blk, 0, stream>>>(values, Wv, Xv, nullptr, nullptr, NTOK);

    attn_kernel<<<NTOK, blk, 0, stream>>>(Xq, Xk, Xv);   // O overwrites Xq

    gemm_abt_kernel<<<grid, blk, 0, stream>>>(Xq, Wo, out, bo, query, NTOK);
    layernorm_kernel<<<NTOK, blk, 0, stream>>>(out, gamma, beta);

    (void)in_sizes; (void)n_in; (void)out_size; (void)ws_size;
}